// BertCRF_90254442758437
// MI455X (gfx1250) — compile-verified
//
#include <hip/hip_runtime.h>
#include <hip/hip_bf16.h>

// ---------------------------------------------------------------------------
// BertCRF on gfx1250 (MI455X, wave32)
//   Kernel 1: logits = hidden @ W + b   via V_WMMA_F32_16X16X4_F32
//             W staged transposed+padded in LDS -> branch-free inner loop
//   Kernel 2: CRF forward (logZ - numerator) per batch, one wave32 per batch
//   Kernel 3: deterministic sum of 64 per-batch partials -> nll
// ---------------------------------------------------------------------------

typedef float v2f __attribute__((ext_vector_type(2)));
typedef float v8f __attribute__((ext_vector_type(8)));

#define BB 64
#define SS 512
#define HH 768
#define KK 21

#define KCHUNK 256               // K-slice staged in LDS per refill
#define WT_PITCH 260             // 256 + 4 pad: conflict-free b64 ds loads

// -------------------------- GEMM: 32768x768 @ 768x21 -----------------------
// Grid: 256 blocks x 256 threads (8 waves). Each wave computes one 16-row
// M-tile across both 16-wide N-tiles (cols 0..15 and 16..31; 21..31 are
// zero-padded in LDS so the inner loop has no bounds checks at all).
__global__ void __launch_bounds__(256)
bertcrf_logits_wmma(const float* __restrict__ H, const float* __restrict__ W,
                    const float* __restrict__ bias, float* __restrict__ out) {
    __shared__ float Wt[32][WT_PITCH];   // [col][k-local], cols >= 21 are zero

    const int lane = threadIdx.x & 31;
    const int wave = threadIdx.x >> 5;
    const int tile = blockIdx.x * 8 + wave;   // 0..2047
    const int row0 = tile * 16;
    const int half = lane >> 4;               // 0: lanes 0-15, 1: lanes 16-31
    const int lr   = lane & 15;

    // A fragment source: row (row0+lr), k-offset 2*half within each 4-wide step
    const float* arow = H + (size_t)(row0 + lr) * HH + half * 2;

    const int colA = lr;        // 0..15  (always < 21)
    const int colB = 16 + lr;   // 16..31 (cols >= 21 read zeros from LDS)

    v8f c0 = {};
    v8f c1 = {};

    for (int kc = 0; kc < HH; kc += KCHUNK) {
        __syncthreads();   // previous chunk fully consumed
        // Cooperative fill: W[k][col] -> Wt[col][k-kc], zero-pad cols 21..31.
        // Consecutive threads read consecutive cols (coalesced on W rows).
        for (int idx = threadIdx.x; idx < 32 * KCHUNK; idx += 256) {
            const int col = idx & 31;
            const int kl  = idx >> 5;
            Wt[col][kl] = (col < KK) ? W[(kc + kl) * KK + col] : 0.0f;
        }
        __syncthreads();

#pragma unroll 4
        for (int kl = 0; kl < KCHUNK; kl += 4) {
            // A: 16x4 f32 tile, 2 VGPRs per lane (ISA 7.12.2 layout)
            v2f a  = *(const v2f*)(arow + kc + kl);
            // B: 4x16 tiles, contiguous pair {K, K+1} per lane from LDS
            v2f b0 = *(const v2f*)(&Wt[colA][kl + half * 2]);
            v2f b1 = *(const v2f*)(&Wt[colB][kl + half * 2]);

            c0 = __builtin_amdgcn_wmma_f32_16x16x4_f32(false, a, false, b0,
                                                       (short)0, c0, false, false);
            c1 = __builtin_amdgcn_wmma_f32_16x16x4_f32(false, a, false, b1,
                                                       (short)0, c1, false, false);
        }
    }

    // D layout: VGPR r -> M = r + 8*half, N = lane&15 (+16 for second tile)
    const bool  bOK   = (colB < KK);
    const float biasA = bias[colA];
    const float biasB = bOK ? bias[colB] : 0.0f;
#pragma unroll
    for (int r = 0; r < 8; ++r) {
        const int row = row0 + r + half * 8;
        out[(size_t)row * KK + colA] = c0[r] + biasA;
        if (bOK) out[(size_t)row * KK + colB] = c1[r] + biasB;
    }
}

// ------------------------------ CRF forward --------------------------------
// Grid: 64 blocks x 32 threads (one wave32 per batch). Lane j owns tag j.
__global__ void __launch_bounds__(32)
bertcrf_crf(const float* __restrict__ logits, const int* __restrict__ labels,
            const unsigned char* __restrict__ mask,
            const float* __restrict__ start_t, const float* __restrict__ trans,
            const float* __restrict__ end_t, float* __restrict__ partial) {
    const int b = blockIdx.x;
    const int j = threadIdx.x;   // 0..31
    const float NEG = -3.0e38f;

    __shared__ float alpha_s[32];
    __shared__ float trans_s[KK * KK];

    for (int idx = j; idx < KK * KK; idx += 32) trans_s[idx] = trans[idx];

    const float*         lg = logits + (size_t)b * SS * KK;
    const int*           lb = labels + b * SS;
    const unsigned char* mk = mask + b * SS;

    float alpha = (j < KK) ? (start_t[j] + lg[j]) : NEG;
    alpha_s[j] = alpha;
    __syncthreads();

    for (int t = 1; t < SS; ++t) {
        float na = alpha;
        if (j < KK) {
            const float em = lg[t * KK + j];
            float v[KK];
            float m = NEG;
#pragma unroll
            for (int i = 0; i < KK; ++i) {
                v[i] = alpha_s[i] + trans_s[i * KK + j];
                m = fmaxf(m, v[i]);
            }
            float s = 0.0f;
#pragma unroll
            for (int i = 0; i < KK; ++i) s += __expf(v[i] - m);
            const float nxt = m + __logf(s) + em;
            if (mk[t]) na = nxt;
        }
        __syncthreads();            // all reads of alpha_s done
        alpha = na;
        alpha_s[j] = alpha;
        __syncthreads();            // writes visible
    }

    // logZ = logsumexp_j(alpha[j] + end_t[j]) via wave32 shuffle reduction
    float val = (j < KK) ? (alpha + end_t[j]) : NEG;
    float m = val;
#pragma unroll
    for (int off = 16; off >= 1; off >>= 1)
        m = fmaxf(m, __shfl_xor(m, off, 32));
    float s = __expf(val - m);
#pragma unroll
    for (int off = 16; off >= 1; off >>= 1)
        s += __shfl_xor(s, off, 32);
    const float logZ = m + __logf(s);

    if (j == 0) {
        // numerator (gold-path score)
        int len = 0;
        for (int t = 0; t < SS; ++t) len += (int)mk[t];
        float num = start_t[lb[0]] + lg[lb[0]];
        for (int t = 1; t < SS; ++t) {
            if (mk[t]) num += lg[t * KK + lb[t]] + trans_s[lb[t - 1] * KK + lb[t]];
        }
        num += end_t[lb[len - 1]];
        partial[b] = logZ - num;
    }
}

// --------------------------- final reduction -------------------------------
__global__ void bertcrf_reduce(const float* __restrict__ partial,
                               float* __restrict__ out) {
    if (threadIdx.x == 0) {
        float s = 0.0f;
        for (int i = 0; i < BB; ++i) s += partial[i];
        out[(size_t)BB * SS * KK] = s;   // nll slot after the logits
    }
}

extern "C" void kernel_launch(void* const* d_in, const int* in_sizes, int n_in,
                              void* d_out, int out_size, void* d_ws, size_t ws_size,
                              hipStream_t stream) {
    (void)in_sizes; (void)n_in; (void)out_size; (void)ws_size;
    const float*         hidden  = (const float*)d_in[0];
    const float*         W       = (const float*)d_in[1];
    const float*         bias    = (const float*)d_in[2];
    const float*         start_t = (const float*)d_in[3];
    const float*         trans   = (const float*)d_in[4];
    const float*         end_t   = (const float*)d_in[5];
    const int*           labels  = (const int*)d_in[6];
    const unsigned char* mask    = (const unsigned char*)d_in[7];

    float* out     = (float*)d_out;      // [B*S*K logits][1 nll]
    float* partial = (float*)d_ws;       // 64 floats

    // 2048 M-tiles of 16 rows, 8 waves per block -> 256 blocks
    bertcrf_logits_wmma<<<256, 256, 0, stream>>>(hidden, W, bias, out);
    bertcrf_crf<<<BB, 32, 0, stream>>>(out, labels, mask, start_t, trans,
                                       end_t, partial);
    bertcrf_reduce<<<1, 32, 0, stream>>>(partial, out);
}